// BilinearSampler_16836271800603
// MI455X (gfx1250) — compile-verified
//
#include <hip/hip_runtime.h>
#include <hip/hip_bf16.h>

// Problem constants (from reference setup_inputs):
// B=8, N=32768, C=128 per plane, R=128, padding=0 -> denom = 1.001
// out: (B, N, 384) f32, channel k = plane*128 + ch, planes ordered xz, xy, yz.
constexpr int kB = 8;
constexpr int kN = 32768;            // points per batch
constexpr int kC = 128;              // channels per plane
constexpr int kR = 128;              // plane resolution
constexpr int kPlaneHW = kR * kR;    // 16384 elements per channel slice
constexpr int kOutC = 3 * kC;        // 384
constexpr int kPtsPerBlock = 64;     // points handled per block
constexpr int kPtsPerWave = 8;       // 8 waves/block * 8 pts/wave = 64

typedef float        v4f __attribute__((ext_vector_type(4)));
typedef unsigned int v4u __attribute__((ext_vector_type(4)));
typedef int          v4i __attribute__((ext_vector_type(4)));
typedef int          v8i __attribute__((ext_vector_type(8)));

#if defined(__has_builtin)
#if __has_builtin(__builtin_amdgcn_tensor_load_to_lds) && \
    __has_builtin(__builtin_amdgcn_s_wait_tensorcnt)
#define USE_TDM 1
#endif
#endif

__global__ __launch_bounds__(256) void triplane_sample_kernel(
    const float* __restrict__ p,
    const float* __restrict__ c_xz,
    const float* __restrict__ c_xy,
    const float* __restrict__ c_yz,
    float* __restrict__ out)
{
    __shared__ float sp[kPtsPerBlock * 3];      // 768 B coordinate tile

    const int lane  = threadIdx.x & 31;
    const int wave  = threadIdx.x >> 5;
    const int plane = blockIdx.y;               // scalar: 0=xz, 1=xy, 2=yz
    const int pt0   = blockIdx.x * kPtsPerBlock;

#if defined(USE_TDM)
    // --- Tensor Data Mover: DMA the block's 64x3-float coord tile to LDS ---
    // D# per CDNA5 ISA: group0 {count=1, lds_addr=0,
    // global_addr, type=2}; group1 {data_size=4B, tensor_dim0=192,
    // tensor_dim1=1, tile_dim0=192, tile_dim1=1, stride0=192}; groups 2/3 zero.
    if (wave == 0) {
        const unsigned long long ga =
            (unsigned long long)(const void*)p + (unsigned long long)pt0 * 12ull;
        v4u g0 = { 1u,                                   // count=1, user mode
                   0u,                                   // lds_addr (bytes) = 0
                   (unsigned)(ga & 0xffffffffu),         // global_addr[31:0]
                   (unsigned)((ga >> 32) | (2u << 30)) };// addr[56:32] | type=2
        v8i g1 = { (int)(2u << 16),      // workgroup_mask=0, data_size=4B
                   (int)(192u << 16),    // tensor_dim0[15:0]=192 in bits[63:48]
                   (int)(1u << 16),      // dim0 hi=0, tensor_dim1 lo=1
                   (int)(192u << 16),    // tensor_dim1 hi=0, tile_dim0=192
                   1,                    // tile_dim1=1, tile_dim2=0
                   192,                  // tensor_dim0_stride[31:0]=192
                   0, 0 };               // stride0 hi, tensor_dim1_stride=0
        v4i gz4 = { 0, 0, 0, 0 };
        v8i gz8 = { 0, 0, 0, 0, 0, 0, 0, 0 };
        // 6-arg variant on this toolchain: (g0, g1, g2, g3, extra, cpol)
        __builtin_amdgcn_tensor_load_to_lds(g0, g1, gz4, gz4, gz8, 0);
        __builtin_amdgcn_s_wait_tensorcnt(0);   // TDM done -> LDS tile valid
    }
    // The TDM engine writes `sp` behind the optimizer's back. Escape the LDS
    // pointer through opaque asm with a memory clobber so GlobalOpt cannot
    // treat the never-(visibly-)stored LDS tile as undef and DCE the kernel.
    {
        void* spp = (void*)sp;
        asm volatile("" : "+r"(spp) : : "memory");
    }
#else
    // Fallback: cooperative staged load of the coordinate tile
    for (int t = threadIdx.x; t < kPtsPerBlock * 3; t += 256)
        sp[t] = p[(size_t)pt0 * 3 + t];
#endif
    __syncthreads();

    // --- scalar (wave-uniform) setup ------------------------------------
    const int b = pt0 >> 15;                    // whole block is in one batch
    const float* planeBase =
        ((plane == 0) ? c_xz : (plane == 1) ? c_xy : c_yz)
        + (size_t)b * kC * kPlaneHW;
    const int d0 = (plane == 2) ? 1 : 0;                       // feeds x (W)
    const int d1 = (plane == 0) ? 2 : ((plane == 1) ? 1 : 2);  // feeds y (H)
    const int ch0 = lane * 4;                   // lane's 4 channels
    const float* chBase = planeBase + (size_t)ch0 * kPlaneHW;

#pragma unroll 2
    for (int j = 0; j < kPtsPerWave; ++j) {
        const int lp = wave * kPtsPerWave + j;  // local point id (wave-uniform)
        const int pt = pt0 + lp;

        // LDS broadcast reads of this point's two coordinates
        const float cu = sp[lp * 3 + d0];
        const float cv = sp[lp * 3 + d1];

        // normalize: p/1.001 + 0.5, clip [0, 1-1e-5]; clamps on x0/x1 provably no-ops
        float u = cu / 1.001f + 0.5f;
        float v = cv / 1.001f + 0.5f;
        u = fminf(fmaxf(u, 0.0f), 1.0f - 1e-5f);
        v = fminf(fmaxf(v, 0.0f), 1.0f - 1e-5f);
        const float x = u * (float)(kR - 1);    // <= 126.998 -> x1 = x0+1
        const float y = v * (float)(kR - 1);
        const float x0f = floorf(x);
        const float y0f = floorf(y);
        const float wx = x - x0f;
        const float wy = y - y0f;
        const int x0 = (int)x0f;                // in [0,126]
        const int y0 = (int)y0f;

        const float w00 = (1.0f - wx) * (1.0f - wy);
        const float w01 = wx * (1.0f - wy);
        const float w10 = (1.0f - wx) * wy;
        const float w11 = wx * wy;

        // gather 4 corners x 4 channels (L2-resident plane data)
        const float* r = chBase + (size_t)(y0 * kR + x0);
        v4f acc;
#pragma unroll
        for (int c = 0; c < 4; ++c) {
            const float* r0 = r + (size_t)c * kPlaneHW;   // row y0: v00,v01
            const float* r1 = r0 + kR;                    // row y1: v10,v11
            const float v00 = r0[0];
            const float v01 = r0[1];
            const float v10 = r1[0];
            const float v11 = r1[1];
            acc[c] = fmaf(v11, w11, fmaf(v10, w10, fmaf(v01, w01, v00 * w00)));
        }

        // streaming output: non-temporal b128 store (don't evict the planes
        // from the 192MB L2 — output is write-once, 402MB total)
        float* o = out + (size_t)pt * kOutC + plane * kC + ch0;
        __builtin_nontemporal_store(acc, (v4f*)o);
    }
}

extern "C" void kernel_launch(void* const* d_in, const int* in_sizes, int n_in,
                              void* d_out, int out_size, void* d_ws, size_t ws_size,
                              hipStream_t stream) {
    const float* p    = (const float*)d_in[0];   // (B, N, 3)
    const float* c_xz = (const float*)d_in[1];   // (B, C, R, R)
    const float* c_xy = (const float*)d_in[2];
    const float* c_yz = (const float*)d_in[3];
    float* out = (float*)d_out;                  // (B, N, 384)

    dim3 grid(kB * kN / kPtsPerBlock, 3);        // (4096, 3)
    triplane_sample_kernel<<<grid, 256, 0, stream>>>(p, c_xz, c_xy, c_yz, out);
}